// VectorQuantizer_44169443672296
// MI455X (gfx1250) — compile-verified
//
#include <hip/hip_runtime.h>

typedef __attribute__((ext_vector_type(2))) float v2f;
typedef __attribute__((ext_vector_type(8))) float v8f;

#define T_DIM 16384
#define D_DIM 5
#define B_DIM 32
#define K_DIM 256
#define DT    (D_DIM * T_DIM)            // 81920
#define NPTS  (B_DIM * T_DIM)            // 524288 points
#define NOUT  (B_DIM * D_DIM * T_DIM)    // 2621440 output elements
#define WAVES_PER_BLOCK 8
#define GROUPS_PER_WAVE 8
#define GROUPS_PER_BLOCK (WAVES_PER_BLOCK * GROUPS_PER_WAVE)   // 64 -> 1024 pts/block
#define NBLOCKS (NPTS / (16 * GROUPS_PER_BLOCK))               // 512

// score_k = -2 x.e_k + |e_k|^2  ==  argmin-equivalent to full L2 distance.
// Augmented GEMM: A = [x0..x4, 1, 0, 0] (16x8), B = [-2e; |e|^2; 0; 0] (8x256),
// done as two V_WMMA_F32_16X16X4_F32 per 16x16 tile.

__global__ __launch_bounds__(256) void vq_main(const float* __restrict__ x,
                                               const float* __restrict__ E,
                                               float* __restrict__ out,
                                               float* __restrict__ partial) {
  const int tid  = threadIdx.x;
  const int lane = tid & 31;
  const int wave = tid >> 5;
  const int col  = lane & 15;   // N position within a 16-wide tile
  const int half = lane >> 4;   // which K-pair half (A/B f32 layout)

  __shared__ int   s_idx[WAVES_PER_BLOCK][16];
  __shared__ float s_red[256];
  // Per-lane WMMA B operands for all 16 codeword tiles, staged in LDS so they
  // don't pin 64 VGPRs across the whole kernel.
  __shared__ v2f sB1[16][32];
  __shared__ v2f sB2[16][32];

  // ---- build B operand tables once per block ----
  // 32-bit B 4x16 layout (mirrors A): VGPR0 = K0 (lanes 0-15) / K2 (lanes 16-31),
  //                                   VGPR1 = K1 / K3.
  for (int i = tid; i < 16 * 32; i += 256) {
    int n      = i >> 5;
    int lane_i = i & 31;
    int col_i  = lane_i & 15;
    int half_i = lane_i >> 4;
    int cc     = n * 16 + col_i;
    float e0 = E[0 * K_DIM + cc];
    float e1 = E[1 * K_DIM + cc];
    float e2 = E[2 * K_DIM + cc];
    float e3 = E[3 * K_DIM + cc];
    float e4 = E[4 * K_DIM + cc];
    float nrm = e0 * e0 + e1 * e1 + e2 * e2 + e3 * e3 + e4 * e4;
    v2f b1, b2;
    if (half_i == 0) {
      b1.x = -2.f * e0; b1.y = -2.f * e1;   // rows K=0,1
      b2.x = -2.f * e4; b2.y = nrm;         // rows K=4,5
    } else {
      b1.x = -2.f * e2; b1.y = -2.f * e3;   // rows K=2,3
      b2.x = 0.f;       b2.y = 0.f;         // rows K=6,7 (zero pad)
    }
    sB1[n][lane_i] = b1;
    sB2[n][lane_i] = b2;
  }
  __syncthreads();   // B tables visible to all waves; last block barrier
                     // until the final SSE reduction.

  float sse = 0.f;

  for (int g = 0; g < GROUPS_PER_WAVE; ++g) {
    const int p_base =
        (blockIdx.x * GROUPS_PER_BLOCK + wave * GROUPS_PER_WAVE + g) * 16;

    // ---- A operand: point p = p_base + col, x is [B, D, T] ----
    int p  = p_base + col;
    int b  = p >> 14;             // T = 16384
    int t  = p & (T_DIM - 1);
    int xb = b * DT + t;
    int xo = xb + half * 2 * T_DIM;
    v2f a1; a1.x = x[xo]; a1.y = x[xo + T_DIM];   // K = 0,1 | 2,3
    // branchless a2: load on all lanes (always in-bounds), mask the upper half
    float x4 = x[xb + 4 * T_DIM];
    v2f a2;
    a2.x = (half == 0) ? x4  : 0.f;               // K = 4 | 6
    a2.y = (half == 0) ? 1.f : 0.f;               // K = 5 | 7

    if (g + 1 < GROUPS_PER_WAVE) {                // next group is p+16: prefetch
      __builtin_prefetch(&x[xo + 16], 0, 3);
      __builtin_prefetch(&x[xo + T_DIM + 16], 0, 3);
      __builtin_prefetch(&x[xb + 4 * T_DIM + 16], 0, 3);
    }

    float runmin[8];
    int   runidx[8];

    // ---- tile 0 initializes (min, idx): no FLT_MAX canonicalization ops ----
    {
      v2f b1 = sB1[0][lane];
      v2f b2 = sB2[0][lane];
      v8f c = {};
      c = __builtin_amdgcn_wmma_f32_16x16x4_f32(false, a1, false, b1,
                                                (short)0, c, false, false);
      c = __builtin_amdgcn_wmma_f32_16x16x4_f32(false, a2, false, b2,
                                                (short)0, c, false, false);
#pragma unroll
      for (int v = 0; v < 8; ++v) { runmin[v] = c[v]; runidx[v] = col; }
    }

    // ---- remaining 15 codeword tiles; per-lane-slot running argmin ----
#pragma unroll
    for (int n = 1; n < 16; ++n) {
      v2f b1 = sB1[n][lane];
      v2f b2 = sB2[n][lane];
      v8f c = {};
      c = __builtin_amdgcn_wmma_f32_16x16x4_f32(false, a1, false, b1,
                                                (short)0, c, false, false);
      c = __builtin_amdgcn_wmma_f32_16x16x4_f32(false, a2, false, b2,
                                                (short)0, c, false, false);
      int idxn = n * 16 + col;
#pragma unroll
      for (int v = 0; v < 8; ++v) {
        bool lt = c[v] < runmin[v];          // strict <: lowest index wins ties
        runmin[v] = lt ? c[v] : runmin[v];
        runidx[v] = lt ? idxn : runidx[v];
      }
    }

    // ---- cross-lane reduce over the 16 column slots (stays within halves) ----
#pragma unroll
    for (int s = 1; s < 16; s <<= 1) {
#pragma unroll
      for (int v = 0; v < 8; ++v) {
        float om = __shfl_xor(runmin[v], s, 32);
        int   oi = __shfl_xor(runidx[v], s, 32);
        bool take = (om < runmin[v]) || ((om == runmin[v]) && (oi < runidx[v]));
        runmin[v] = take ? om : runmin[v];
        runidx[v] = take ? oi : runidx[v];
      }
    }

    // C layout: vgpr v holds point p_base+v (lanes 0-15) / p_base+8+v (16-31).
    // s_idx[wave][*] is private to this wave and DS ops from one wave complete
    // in order, so a compile-time wave fence is sufficient (no block barrier).
    if (col == 0) {
#pragma unroll
      for (int v = 0; v < 8; ++v) s_idx[wave][half * 8 + v] = runidx[v];
    }
    __builtin_amdgcn_wave_barrier();

    // ---- gather winning codeword, write out, accumulate SSE ----
    // Both half-waves participate: lanes 0-15 handle d=0..2, lanes 16-31 d=3..4.
    {
      int k  = s_idx[wave][col];
      int ob = xb;                       // same point p_base+col as above
      int dstart = half ? 3 : 0;
      int dcount = half ? 2 : 3;
#pragma unroll
      for (int j = 0; j < 3; ++j) {
        if (j < dcount) {
          int d = dstart + j;
          float ev = E[d * K_DIM + k];
          out[ob + d * T_DIM] = ev;            // straight-through forward == q
          float dv = x[ob + d * T_DIM] - ev;   // x re-read: L1/L2 hit
          sse += dv * dv;
        }
      }
    }
    __builtin_amdgcn_wave_barrier();   // keep next iter's s_idx write ordered
  }

  // ---- deterministic block reduction of SSE ----
  __syncthreads();
  s_red[tid] = sse;
  __syncthreads();
  for (int k = 128; k > 0; k >>= 1) {
    if (tid < k) s_red[tid] += s_red[tid + k];
    __syncthreads();
  }
  if (tid == 0) partial[blockIdx.x] = s_red[0];
}

__global__ __launch_bounds__(256) void vq_finalize(const float* __restrict__ partial,
                                                   float* __restrict__ out) {
  __shared__ float sm[256];
  float s = 0.f;
  for (int i = threadIdx.x; i < NBLOCKS; i += 256) s += partial[i];
  sm[threadIdx.x] = s;
  __syncthreads();
  for (int k = 128; k > 0; k >>= 1) {
    if (threadIdx.x < k) sm[threadIdx.x] += sm[threadIdx.x + k];
    __syncthreads();
  }
  if (threadIdx.x == 0) {
    float mse = sm[0] / (float)NOUT;
    out[NOUT]     = mse;   // dictionary_loss  (forward value)
    out[NOUT + 1] = mse;   // commitment_loss  (identical forward value)
  }
}

extern "C" void kernel_launch(void* const* d_in, const int* in_sizes, int n_in,
                              void* d_out, int out_size, void* d_ws, size_t ws_size,
                              hipStream_t stream) {
  const float* x = (const float*)d_in[0];   // [32, 5, 16384] f32
  const float* E = (const float*)d_in[1];   // [5, 256] f32
  float* out     = (float*)d_out;           // [2621440 q | 1 dict | 1 commit]
  float* partial = (float*)d_ws;            // NBLOCKS floats of scratch

  vq_main<<<NBLOCKS, 256, 0, stream>>>(x, E, out, partial);
  vq_finalize<<<1, 256, 0, stream>>>(partial, out);
}